// CrossBandAttention_24567212933356
// MI455X (gfx1250) — compile-verified
//
#include <hip/hip_runtime.h>
#include <math.h>

// Problem constants (B=4, C=3, H=W=256, E=64, HEADS=8, HD=8)
#define HW    65536      // H*W
#define NPIX  262144     // B*H*W
#define CHW   196608     // C*H*W
#define BCHW  786432     // B*C*H*W

typedef __attribute__((ext_vector_type(2))) float v2f;
typedef __attribute__((ext_vector_type(8))) float v8f;

// workspace layout (float offsets)
#define WS_COMBQKV 0      // [3][192][4]  folded (in_proj @ Wp), k-padded to 4
#define WS_CBIAS   2304   // [3][192]     in_proj_w@bp + in_proj_b
#define WS_COMBOT  2880   // [3][16][64]  folded (Wo @ out_proj_w)^T-ish, n-padded to 16
#define WS_OBIAS   5952   // [3][16]      Wo@out_proj_b + bo, n-padded
#define WS_GATEW   6000   // [4]          softmax(gates)
#define WS_TOTAL   6004

// ---------------------------------------------------------------------------
// Precompute: fold the linear maps (tiny, runs every launch -> deterministic)
// ---------------------------------------------------------------------------
__global__ void cba_precompute(const float* __restrict__ Wp,  const float* __restrict__ bp,
                               const float* __restrict__ ipw, const float* __restrict__ ipb,
                               const float* __restrict__ opw, const float* __restrict__ opb,
                               const float* __restrict__ Wo,  const float* __restrict__ bo,
                               const float* __restrict__ gates, float* __restrict__ ws) {
  int t = blockIdx.x * blockDim.x + threadIdx.x;
  if (t < 2304) {                       // combQKV[band][j][c] = sum_e ipw[j][e]*Wp[band][e][c]
    int band = t / 768, r = t % 768, j = r >> 2, c = r & 3;
    float acc = 0.f;
    if (c < 3)
      for (int e = 0; e < 64; ++e)
        acc += ipw[j * 64 + e] * Wp[(band * 64 + e) * 3 + c];
    ws[WS_COMBQKV + t] = acc;
  } else if (t < 2880) {                // cbias[band][j] = ipw[j]@bp[band] + ipb[j]
    int i = t - 2304, band = i / 192, j = i % 192;
    float acc = ipb[j];
    for (int e = 0; e < 64; ++e) acc += ipw[j * 64 + e] * bp[band * 64 + e];
    ws[t] = acc;
  } else if (t < 5952) {                // combOT[band][n][e] = sum_e2 Wo[band][n][e2]*opw[e2][e]
    int i = t - 2880, band = i >> 10, r = i & 1023, n = r >> 6, e = r & 63;
    float acc = 0.f;
    if (n < 3)
      for (int e2 = 0; e2 < 64; ++e2)
        acc += Wo[(band * 3 + n) * 64 + e2] * opw[e2 * 64 + e];
    ws[t] = acc;
  } else if (t < 6000) {                // obias[band][n] = Wo[band][n]@opb + bo[band][n]
    int i = t - 5952, band = i >> 4, n = i & 15;
    float acc = 0.f;
    if (n < 3) {
      acc = bo[band * 3 + n];
      for (int e2 = 0; e2 < 64; ++e2) acc += Wo[(band * 3 + n) * 64 + e2] * opb[e2];
    }
    ws[t] = acc;
  } else if (t < WS_TOTAL) {            // gate softmax
    int band = t - 6000;
    float g0 = gates[0], g1 = gates[1], g2 = gates[2];
    float m = fmaxf(g0, fmaxf(g1, g2));
    float e0 = __expf(g0 - m), e1 = __expf(g1 - m), e2 = __expf(g2 - m);
    float inv = 1.0f / (e0 + e1 + e2);
    float val = 0.f;
    if (band == 0) val = e0 * inv;
    else if (band == 1) val = e1 * inv;
    else if (band == 2) val = e2 * inv;
    ws[t] = val;
  }
}

// ---------------------------------------------------------------------------
// Main fused kernel: one wave32 per 16 pixels.
//  Stage 1: raw[16x3pad4] x combQKV -> qkv (f32 WMMA 16x16x4, 12 tiles/band)
//  Stage 2: 3x3 attention per (pixel, head) on VALU (128 tasks / 4 rounds)
//  Stage 3: o[16x64] x combOT -> out channels (16 chained WMMAs/band) + gated residual
// ---------------------------------------------------------------------------
__launch_bounds__(32)
__global__ void cba_main(const float* __restrict__ b0,
                         const float* __restrict__ b1,
                         const float* __restrict__ b2,
                         const float* __restrict__ ws,
                         float* __restrict__ out) {
  __shared__ float sm[3 * 16 * 192];   // 36 KB: [band][pix][j] qkv, q-slot reused for o
  const int lane = threadIdx.x;        // 0..31
  const int n16  = lane & 15;
  const int half = lane >> 4;
  const int p0   = blockIdx.x * 16;

  const float* combQ  = ws + WS_COMBQKV;
  const float* cbias  = ws + WS_CBIAS;
  const float* combOT = ws + WS_COMBOT;
  const float* obias  = ws + WS_OBIAS;
  const float* gatew  = ws + WS_GATEW;

  // ---- Stage 1: folded QKV via WMMA f32 16x16x4 (K: 3 channels + zero pad)
  const int p   = p0 + n16;
  const int img = p >> 16;             // H*W = 65536
  const int hw  = p & 65535;
  #pragma unroll
  for (int band = 0; band < 3; ++band) {
    const float* src = (band == 0) ? b0 : (band == 1) ? b1 : b2;
    // A layout: lane%16 = row M (pixel), VGPR v holds K = 2*half + v
    v2f a;
    a.x = src[img * CHW + (half * 2) * HW + hw];
    a.y = (half == 0) ? src[img * CHW + HW + hw] : 0.0f;   // c=1 valid, c=3 is pad
    for (int t = 0; t < 12; ++t) {
      int j = t * 16 + n16;
      // B layout: lane%16 = col N, VGPR v holds K = 2*half + v  (contiguous pair)
      const float* bptr = combQ + (band * 192 + j) * 4 + half * 2;
      v2f bm; bm.x = bptr[0]; bm.y = bptr[1];
      float bias = cbias[band * 192 + j];
      v8f c;
      #pragma unroll
      for (int v = 0; v < 8; ++v) c[v] = bias;
      v8f d = __builtin_amdgcn_wmma_f32_16x16x4_f32(
          false, a, false, bm, (short)0, c, false, false);
      // D layout: lane%16 = N (=j), VGPR v = row M (pixel v + 8*half)
      #pragma unroll
      for (int v = 0; v < 8; ++v)
        sm[(band * 16 + (v + 8 * half)) * 192 + j] = d[v];
    }
  }
  __syncthreads();

  // ---- Stage 2: attention, lanes = (pixel, head) tasks
  const float scale = 0.35355339059327373f;   // 1/sqrt(8)
  for (int r = 0; r < 4; ++r) {
    int task = r * 32 + lane;
    int pix  = task >> 3;
    int head = task & 7;
    float q[3][8], k[3][8], vv[3][8];
    #pragma unroll
    for (int band = 0; band < 3; ++band) {
      const float* base = sm + (band * 16 + pix) * 192 + head * 8;
      #pragma unroll
      for (int d = 0; d < 8; ++d) {
        q[band][d]  = base[d];
        k[band][d]  = base[64 + d];
        vv[band][d] = base[128 + d];
      }
    }
    float o[3][8];
    #pragma unroll
    for (int i = 0; i < 3; ++i) {
      float sc[3];
      #pragma unroll
      for (int j = 0; j < 3; ++j) {
        float acc = 0.f;
        #pragma unroll
        for (int d = 0; d < 8; ++d) acc += q[i][d] * k[j][d];
        sc[j] = acc * scale;
      }
      float m  = fmaxf(sc[0], fmaxf(sc[1], sc[2]));
      float e0 = __expf(sc[0] - m), e1 = __expf(sc[1] - m), e2 = __expf(sc[2] - m);
      float inv = 1.0f / (e0 + e1 + e2);
      e0 *= inv; e1 *= inv; e2 *= inv;
      #pragma unroll
      for (int d = 0; d < 8; ++d)
        o[i][d] = e0 * vv[0][d] + e1 * vv[1][d] + e2 * vv[2][d];
    }
    #pragma unroll
    for (int i = 0; i < 3; ++i) {       // overwrite q-slots with o (disjoint per task)
      float* base = sm + (i * 16 + pix) * 192 + head * 8;
      #pragma unroll
      for (int d = 0; d < 8; ++d) base[d] = o[i][d];
    }
  }
  __syncthreads();

  // ---- Stage 3: folded output projection, K=64 chained WMMAs, gated residual
  #pragma unroll
  for (int band = 0; band < 3; ++band) {
    v8f c;
    float bias = obias[band * 16 + n16];
    #pragma unroll
    for (int v = 0; v < 8; ++v) c[v] = bias;
    #pragma unroll
    for (int k0 = 0; k0 < 64; k0 += 4) {
      const float* aptr = sm + (band * 16 + n16) * 192 + k0 + half * 2;   // o features
      v2f a;  a.x = aptr[0]; a.y = aptr[1];
      const float* bptr = combOT + (band * 16 + n16) * 64 + k0 + half * 2;
      v2f bm; bm.x = bptr[0]; bm.y = bptr[1];
      c = __builtin_amdgcn_wmma_f32_16x16x4_f32(
          false, a, false, bm, (short)0, c, false, false);
    }
    // D: lane%16 = out channel (only 0..2 valid), VGPR v = pixel v + 8*half
    float w = gatew[band];
    if (n16 < 3) {
      const float* src = (band == 0) ? b0 : (band == 1) ? b1 : b2;
      #pragma unroll
      for (int v = 0; v < 8; ++v) {
        int pp  = p0 + v + 8 * half;
        int im  = pp >> 16;
        int hw2 = pp & 65535;
        int off = im * CHW + n16 * HW + hw2;
        out[band * BCHW + off] = src[off] + w * c[v];
      }
    }
  }
}

extern "C" void kernel_launch(void* const* d_in, const int* in_sizes, int n_in,
                              void* d_out, int out_size, void* d_ws, size_t ws_size,
                              hipStream_t stream) {
  const float* b0    = (const float*)d_in[0];
  const float* b1    = (const float*)d_in[1];
  const float* b2    = (const float*)d_in[2];
  const float* Wp    = (const float*)d_in[3];
  const float* bp    = (const float*)d_in[4];
  const float* ipw   = (const float*)d_in[5];
  const float* ipb   = (const float*)d_in[6];
  const float* opw   = (const float*)d_in[7];
  const float* opb   = (const float*)d_in[8];
  const float* Wo    = (const float*)d_in[9];
  const float* bo    = (const float*)d_in[10];
  const float* gates = (const float*)d_in[11];
  float* ws  = (float*)d_ws;
  float* out = (float*)d_out;

  cba_precompute<<<(WS_TOTAL + 255) / 256, 256, 0, stream>>>(
      Wp, bp, ipw, ipb, opw, opb, Wo, bo, gates, ws);
  cba_main<<<NPIX / 16, 32, 0, stream>>>(b0, b1, b2, ws, out);
}